// FasterRcnnRegionDetector_43233140802196
// MI455X (gfx1250) — compile-verified
//
#include <hip/hip_runtime.h>
#include <hip/hip_bf16.h>

// ---------------- problem constants (match reference) ----------------
#define BIMG   8
#define NPROP  2000
#define CNUM   91            // classes incl background
#define CM1    90            // foreground classes
#define FDIM   1024
#define NCAND  (NPROP * CM1) // 180000 candidates per image
#define KPRE   2048
#define MAXDET 100
#define SCORE_THRESH 0.05f
#define NMS_THRESH   0.5f
#define MIN_SIZE     0.01f
#define BBOX_CLIP    4.135166556742356f   // log(1000/16)

typedef __attribute__((ext_vector_type(2))) float v2f;
typedef __attribute__((ext_vector_type(8))) float v8f;

// ---------------- box decode + clip (shared by two kernels) ----------------
__device__ inline void decode_clip(const float* rg,
                                   float px1, float py1, float px2, float py2,
                                   float W, float H,
                                   float& x1, float& y1, float& x2, float& y2) {
  float w  = px2 - px1;
  float h  = py2 - py1;
  float cx = px1 + 0.5f * w;
  float cy = py1 + 0.5f * h;
  float dx = rg[0] * 0.1f;                     // / WX
  float dy = rg[1] * 0.1f;                     // / WY
  float dw = fminf(rg[2] * 0.2f, BBOX_CLIP);   // / WW
  float dh = fminf(rg[3] * 0.2f, BBOX_CLIP);   // / WH
  float pcx = dx * w + cx;
  float pcy = dy * h + cy;
  float pw  = expf(dw) * w;
  float ph  = expf(dh) * h;
  x1 = fminf(fmaxf(pcx - 0.5f * pw, 0.0f), W);
  y1 = fminf(fmaxf(pcy - 0.5f * ph, 0.0f), H);
  x2 = fminf(fmaxf(pcx + 0.5f * pw, 0.0f), W);
  y2 = fminf(fmaxf(pcy + 0.5f * ph, 0.0f), H);
}

// ---------------- K0: init workspace ----------------
__global__ void k_init(unsigned* hist, unsigned* cnts, int* tinfo,
                       float* sel_s, int* sel_i) {
  int i = blockIdx.x * blockDim.x + threadIdx.x;   // 16384 threads
  if (i < BIMG * 1024) hist[i] = 0u;
  if (i < 16)          cnts[i] = 0u;
  if (i < BIMG * 4)    tinfo[i] = 0;
  if (i < BIMG * KPRE) { sel_s[i] = -1.0f; sel_i[i] = 0x7fffffff; }
}

// ---------------- K1: softmax + decode + mask + score histogram ----------------
// one wave (32 threads) per proposal row
__global__ void k_score(const float* __restrict__ logits,
                        const float* __restrict__ reg,
                        const float* __restrict__ props,
                        const int*   __restrict__ sizes,
                        float* __restrict__ cand, unsigned* __restrict__ hist) {
  int rowg = blockIdx.x;                 // 0 .. B*N-1
  int b    = rowg / NPROP;
  int row  = rowg % NPROP;
  int tid  = threadIdx.x;                // 0..31
  const float* L = logits + (size_t)rowg * CNUM;

  float m = -3.0e38f;
  for (int c = tid; c < CNUM; c += 32) m = fmaxf(m, L[c]);
  for (int o = 16; o > 0; o >>= 1) m = fmaxf(m, __shfl_xor(m, o, 32));
  float s = 0.0f;
  for (int c = tid; c < CNUM; c += 32) s += expf(L[c] - m);
  for (int o = 16; o > 0; o >>= 1) s += __shfl_xor(s, o, 32);
  float inv = 1.0f / s;

  float Hf = (float)sizes[b * 2 + 0];
  float Wf = (float)sizes[b * 2 + 1];
  const float* p = props + (size_t)rowg * 4;
  float px1 = p[0], py1 = p[1], px2 = p[2], py2 = p[3];

  for (int c = tid + 1; c < CNUM; c += 32) {        // foreground classes
    float score = expf(L[c] - m) * inv;
    const float* rg = reg + (size_t)rowg * (CNUM * 4) + c * 4;
    float x1, y1, x2, y2;
    decode_clip(rg, px1, py1, px2, py2, Wf, Hf, x1, y1, x2, y2);
    float bw = x2 - x1, bh = y2 - y1;
    bool valid = (score > SCORE_THRESH) && (bw >= MIN_SIZE) && (bh >= MIN_SIZE);
    cand[(size_t)b * NCAND + row * CM1 + (c - 1)] = valid ? score : -1.0f;
    if (valid) {
      int bin = min(1023, (int)(score * 1024.0f));
      atomicAdd(&hist[b * 1024 + bin], 1u);
    }
  }
}

// ---------------- K2: find score-threshold bin for top-2048 ----------------
__global__ void k_thresh(const unsigned* __restrict__ hist, int* __restrict__ tinfo) {
  int b = blockIdx.x;
  if (threadIdx.x != 0) return;
  const unsigned* h = hist + b * 1024;
  unsigned acc = 0; int T = -1; unsigned above = 0;
  for (int bin = 1023; bin >= 0; --bin) {
    unsigned c = h[bin];
    if (acc + c >= KPRE) { T = bin; above = acc; break; }
    acc += c;
  }
  tinfo[b * 4 + 0] = T;
  tinfo[b * 4 + 1] = (int)above;
  tinfo[b * 4 + 2] = (T >= 0) ? (KPRE - (int)above) : 0;
}

// ---------------- K3: compact candidates above threshold into 2048 slots ----------------
__global__ void k_compact(const float* __restrict__ cand,
                          const int* __restrict__ tinfo, unsigned* __restrict__ cnts,
                          float* __restrict__ sel_s, int* __restrict__ sel_i) {
  int g = blockIdx.x * blockDim.x + threadIdx.x;
  if (g >= BIMG * NCAND) return;
  int b = g / NCAND;
  int i = g % NCAND;
  float s = cand[g];
  if (s <= 0.0f) return;                       // invalid / below thresh
  int bin   = min(1023, (int)(s * 1024.0f));
  int T     = tinfo[b * 4 + 0];
  int above = tinfo[b * 4 + 1];
  int slots = tinfo[b * 4 + 2];
  if (bin > T) {
    unsigned pos = atomicAdd(&cnts[b], 1u);          // pos < above <= 2047
    sel_s[b * KPRE + pos] = s;
    sel_i[b * KPRE + pos] = i;
  } else if (bin == T) {
    unsigned q = atomicAdd(&cnts[8 + b], 1u);
    if ((int)q < slots) {
      int pos = above + (int)q;
      sel_s[b * KPRE + pos] = s;
      sel_i[b * KPRE + pos] = i;
    }
  }
}

// ---------------- K4: bitonic sort 2048 (desc score, asc idx) + re-decode boxes ----------------
__global__ void k_sort(const float* __restrict__ sel_s, const int* __restrict__ sel_i,
                       const float* __restrict__ reg, const float* __restrict__ props,
                       const int* __restrict__ sizes,
                       float* __restrict__ srt_s, float* __restrict__ cbx,
                       int* __restrict__ lab, int* __restrict__ frow) {
  __shared__ float ss[KPRE];
  __shared__ int   ii[KPRE];
  int b = blockIdx.x, tid = threadIdx.x;            // 1024 threads
  ss[tid]        = sel_s[b * KPRE + tid];
  ii[tid]        = sel_i[b * KPRE + tid];
  ss[tid + 1024] = sel_s[b * KPRE + tid + 1024];
  ii[tid + 1024] = sel_i[b * KPRE + tid + 1024];
  __syncthreads();
  for (int k = 2; k <= KPRE; k <<= 1) {
    for (int j = k >> 1; j > 0; j >>= 1) {
      for (int i0 = tid; i0 < KPRE; i0 += 1024) {
        int ixj = i0 ^ j;
        if (ixj > i0) {
          float a = ss[i0], c = ss[ixj];
          int  ai = ii[i0], ci = ii[ixj];
          // ranksBefore(x,y): x before y in descending-score / ascending-idx order
          bool lowFirst = (c > a) || (c == a && ci < ai);  // ranksBefore(ixj, i0)
          bool hiFirst  = (a > c) || (a == c && ai < ci);  // ranksBefore(i0, ixj)
          bool doSwap = ((i0 & k) == 0) ? lowFirst : hiFirst;
          if (doSwap) { ss[i0] = c; ss[ixj] = a; ii[i0] = ci; ii[ixj] = ai; }
        }
      }
      __syncthreads();
    }
  }
  float Hf = (float)sizes[b * 2 + 0];
  float Wf = (float)sizes[b * 2 + 1];
  for (int t = tid; t < KPRE; t += 1024) {
    float cs = ss[t];
    int   ci = ii[t];
    float x1 = 0, y1 = 0, x2 = 0, y2 = 0;
    int l = -1, fr = -1;
    if (cs > 0.0f) {
      int row = ci / CM1;
      int cls = ci % CM1 + 1;
      int rowg = b * NPROP + row;
      const float* p = props + (size_t)rowg * 4;
      const float* rg = reg + (size_t)rowg * (CNUM * 4) + cls * 4;
      decode_clip(rg, p[0], p[1], p[2], p[3], Wf, Hf, x1, y1, x2, y2);
      l = cls; fr = row;
    } else {
      cs = -1.0f;
    }
    size_t o = (size_t)b * KPRE + t;
    srt_s[o]  = cs;
    cbx[o * 4 + 0] = x1; cbx[o * 4 + 1] = y1;
    cbx[o * 4 + 2] = x2; cbx[o * 4 + 3] = y2;
    lab[o]  = l;
    frow[o] = fr;
  }
}

// ---------------- K5: suppression bitmask matrix via wave32 ballot ----------------
__global__ void k_supmat(const float* __restrict__ cbx, const int* __restrict__ lab,
                         const int* __restrict__ sizes, unsigned* __restrict__ sup) {
  int i = blockIdx.x;                  // row
  int b = blockIdx.y;                  // image
  const float* base = cbx + (size_t)b * KPRE * 4;
  float Hf = (float)sizes[b * 2 + 0];
  float Wf = (float)sizes[b * 2 + 1];
  float off = fmaxf(Hf, Wf) + 1.0f;
  float oi = (float)lab[b * KPRE + i] * off;
  float ax1 = base[i * 4 + 0] + oi, ay1 = base[i * 4 + 1] + oi;
  float ax2 = base[i * 4 + 2] + oi, ay2 = base[i * 4 + 3] + oi;
  float areaA = (ax2 - ax1) * (ay2 - ay1);
  int lane = threadIdx.x & 31;
  for (int j = threadIdx.x; j < KPRE; j += 256) {
    bool sbit = false;
    if (j > i) {
      float oj = (float)lab[b * KPRE + j] * off;
      float bx1 = base[j * 4 + 0] + oj, by1 = base[j * 4 + 1] + oj;
      float bx2 = base[j * 4 + 2] + oj, by2 = base[j * 4 + 3] + oj;
      float areaB = (bx2 - bx1) * (by2 - by1);
      float ltx = fmaxf(ax1, bx1), lty = fmaxf(ay1, by1);
      float rbx = fminf(ax2, bx2), rby = fminf(ay2, by2);
      float iw = fmaxf(rbx - ltx, 0.0f), ih = fmaxf(rby - lty, 0.0f);
      float inter = iw * ih;
      float iou = inter / (areaA + areaB - inter + 1e-9f);
      sbit = iou > NMS_THRESH;
    }
    unsigned mword = __builtin_amdgcn_ballot_w32(sbit);
    if (lane == 0) sup[((size_t)(b * KPRE + i)) * 64 + (j >> 5)] = mword;
  }
}

// ---------------- K6: sequential greedy suppression over bitmasks ----------------
__global__ void k_suppress(const float* __restrict__ srt_s,
                           const unsigned* __restrict__ sup,
                           unsigned* __restrict__ keep_out) {
  int b = blockIdx.x, tid = threadIdx.x;      // 64 threads
  __shared__ unsigned kw[64];
  unsigned w = 0;
  for (int bit = 0; bit < 32; ++bit) {
    int j = tid * 32 + bit;
    if (srt_s[b * KPRE + j] > 0.0f) w |= (1u << bit);
  }
  kw[tid] = w;
  __syncthreads();
  for (int i = 0; i < KPRE; ++i) {
    bool alive = (kw[i >> 5] >> (i & 31)) & 1u;
    __syncthreads();
    if (alive) kw[tid] &= ~sup[((size_t)(b * KPRE + i)) * 64 + tid];
    __syncthreads();
  }
  keep_out[b * 64 + tid] = kw[tid];
}

// ---------------- K7: take first 100 kept (already score-sorted) ----------------
__global__ void k_select(const unsigned* __restrict__ keep,
                         const float* __restrict__ srt_s, const float* __restrict__ cbx,
                         const int* __restrict__ lab, const int* __restrict__ frow,
                         int* __restrict__ fr100,
                         float* __restrict__ out_boxes, float* __restrict__ out_scores,
                         float* __restrict__ out_labels) {
  int b = blockIdx.x, tid = threadIdx.x;     // 256 threads
  __shared__ int list[MAXDET];
  __shared__ int cnt;
  if (tid == 0) {
    int c = 0;
    for (int i = 0; i < KPRE && c < MAXDET; ++i)
      if ((keep[b * 64 + (i >> 5)] >> (i & 31)) & 1u) list[c++] = i;
    cnt = c;
  }
  __syncthreads();
  for (int s = tid; s < 112; s += 256) {
    if (s < MAXDET) {
      if (s < cnt) {
        int c = list[s];
        size_t o = (size_t)b * KPRE + c;
        for (int q = 0; q < 4; ++q)
          out_boxes[((size_t)b * MAXDET + s) * 4 + q] = cbx[o * 4 + q];
        out_scores[b * MAXDET + s] = srt_s[o];
        out_labels[b * MAXDET + s] = (float)lab[o];
        fr100[b * 112 + s] = frow[o];
      } else {
        for (int q = 0; q < 4; ++q)
          out_boxes[((size_t)b * MAXDET + s) * 4 + q] = 0.0f;
        out_scores[b * MAXDET + s] = 0.0f;
        out_labels[b * MAXDET + s] = -1.0f;
        fr100[b * 112 + s] = -1;
      }
    } else {
      fr100[b * 112 + s] = -1;               // pad rows for M-tiling
    }
  }
}

// ---------------- K8: feature gather as exact f32 selection-GEMM on WMMA ----------------
// D[100x1024] = OneHot[100x2000] x Feats[2000x1024], bit-exact in f32.
// K-loop pruned: each M-tile of 16 rows touches at most 16 k-steps of 4.
__global__ void k_feat_wmma(const int* __restrict__ fr100,
                            const float* __restrict__ feats,
                            float* __restrict__ out_feats) {
  int ng = blockIdx.x;        // 0..7 : group of 8 N-tiles (128 cols)
  int mt = blockIdx.y;        // 0..6 : M-tile (16 rows, covers 112 >= 100)
  int b  = blockIdx.z;        // image
  int lane = threadIdx.x;     // wave32

  const int* fr = fr100 + b * 112 + mt * 16;
  int frs[16];
#pragma unroll
  for (int r = 0; r < 16; ++r) frs[r] = fr[r];

  int m = lane & 15;
  int koff = (lane < 16) ? 0 : 2;          // ISA A layout: lanes0-15 K=0/1, lanes16-31 K=2/3
  int frm = fr[m];                          // this lane's output-row source row

  const float* F = feats + (size_t)b * NPROP * FDIM;
  float* OUT = out_feats + (size_t)b * MAXDET * FDIM;

  v8f acc[8] = {};

#pragma unroll
  for (int r = 0; r < 16; ++r) {
    int frr = frs[r];
    if (frr < 0) continue;
    int ks = frr >> 2;
    bool first = true;
#pragma unroll
    for (int pr = 0; pr < 16; ++pr)
      if (pr < r && frs[pr] >= 0 && (frs[pr] >> 2) == ks) first = false;
    if (!first) continue;                  // this k-step already processed

    int k0 = ks * 4 + koff;
    v2f a;
    a.x = (frm == k0)     ? 1.0f : 0.0f;
    a.y = (frm == k0 + 1) ? 1.0f : 0.0f;
    const float* B0 = F + (size_t)k0 * FDIM;
    const float* B1 = F + (size_t)(k0 + 1) * FDIM;
#pragma unroll
    for (int nt = 0; nt < 8; ++nt) {
      int n0 = ng * 128 + nt * 16;
      v2f bb;
      bb.x = B0[n0 + m];
      bb.y = B1[n0 + m];
      acc[nt] = __builtin_amdgcn_wmma_f32_16x16x4_f32(
          false, a, false, bb, (short)0, acc[nt], false, false);
    }
  }

#pragma unroll
  for (int nt = 0; nt < 8; ++nt) {
    int n0 = ng * 128 + nt * 16;
#pragma unroll
    for (int v = 0; v < 8; ++v) {
      int M = v + ((lane < 16) ? 0 : 8);   // ISA C/D layout
      int orow = mt * 16 + M;
      if (orow < MAXDET) OUT[(size_t)orow * FDIM + n0 + m] = acc[nt][v];
    }
  }
}

// ---------------- launch ----------------
extern "C" void kernel_launch(void* const* d_in, const int* in_sizes, int n_in,
                              void* d_out, int out_size, void* d_ws, size_t ws_size,
                              hipStream_t stream) {
  const float* logits = (const float*)d_in[0];
  const float* feats  = (const float*)d_in[1];
  const float* reg    = (const float*)d_in[2];
  const float* props  = (const float*)d_in[3];
  const int*   sizes  = (const int*)d_in[4];

  float* out = (float*)d_out;
  float* out_boxes  = out;                                   // [8,100,4]
  float* out_feats  = out + BIMG * MAXDET * 4;               // [8,100,1024]
  float* out_scores = out_feats + (size_t)BIMG * MAXDET * FDIM; // [8,100]
  float* out_labels = out_scores + BIMG * MAXDET;            // [8,100]

  char* p = (char*)d_ws;
  auto alloc = [&](size_t bytes) -> void* {
    void* r = (void*)p;
    p += (bytes + 255) & ~(size_t)255;
    return r;
  };
  float*    cand   = (float*)   alloc((size_t)BIMG * NCAND * 4);
  unsigned* hist   = (unsigned*)alloc(BIMG * 1024 * 4);
  int*      tinfo  = (int*)     alloc(BIMG * 4 * 4);
  unsigned* cnts   = (unsigned*)alloc(16 * 4);
  float*    sel_s  = (float*)   alloc(BIMG * KPRE * 4);
  int*      sel_i  = (int*)     alloc(BIMG * KPRE * 4);
  float*    srt_s  = (float*)   alloc(BIMG * KPRE * 4);
  float*    cbx    = (float*)   alloc((size_t)BIMG * KPRE * 4 * 4);
  int*      lab    = (int*)     alloc(BIMG * KPRE * 4);
  int*      frow   = (int*)     alloc(BIMG * KPRE * 4);
  unsigned* sup    = (unsigned*)alloc((size_t)BIMG * KPRE * 64 * 4);
  unsigned* keepw  = (unsigned*)alloc(BIMG * 64 * 4);
  int*      fr100  = (int*)     alloc(BIMG * 112 * 4);

  k_init<<<64, 256, 0, stream>>>(hist, cnts, tinfo, sel_s, sel_i);
  k_score<<<BIMG * NPROP, 32, 0, stream>>>(logits, reg, props, sizes, cand, hist);
  k_thresh<<<BIMG, 32, 0, stream>>>(hist, tinfo);
  k_compact<<<(BIMG * NCAND + 255) / 256, 256, 0, stream>>>(cand, tinfo, cnts, sel_s, sel_i);
  k_sort<<<BIMG, 1024, 0, stream>>>(sel_s, sel_i, reg, props, sizes, srt_s, cbx, lab, frow);
  k_supmat<<<dim3(KPRE, BIMG), 256, 0, stream>>>(cbx, lab, sizes, sup);
  k_suppress<<<BIMG, 64, 0, stream>>>(srt_s, sup, keepw);
  k_select<<<BIMG, 256, 0, stream>>>(keepw, srt_s, cbx, lab, frow, fr100,
                                     out_boxes, out_scores, out_labels);
  k_feat_wmma<<<dim3(8, 7, BIMG), 32, 0, stream>>>(fr100, feats, out_feats);
}